// QueryAndExtractEAEModel_44538810859710
// MI455X (gfx1250) — compile-verified
//
#include <hip/hip_runtime.h>
#include <math.h>

#define B_ 16
#define L_ 512
#define D_ 768
#define S_ 254
#define A_ 40
#define E_ 30
#define H_ 512
#define T_ 256
#define EPS_ 1e-12f

typedef __attribute__((ext_vector_type(2))) float v2f;
typedef __attribute__((ext_vector_type(8))) float v8f;

// ---------------------------------------------------------------------------
// Branchless fp32 WMMA tiles (V_WMMA_F32_16X16X4_F32, wave32, 1 wave per tile
// pair). All strides/shapes are template constants so address math folds and
// out-of-range handling is done by clamping ADDRESSES (never branching):
//  - m/n overflow: clamped loads produce garbage only in D rows/cols that the
//    store epilogue never writes (each D[m][n] depends only on A row m / B col
//    n). All clamped addresses stay inside initialized, finite buffers.
//  - k overflow (K%4!=0 tail only): A fragment multiplied by 0/1 mask.
// ISA layouts (CDNA5 ISA):
//   A 16x4 : lane m = lane&15, VGPR v holds K = 2*(lane>>4)+v
//   B 4x16 : lane n = lane&15, mirrored K arrangement
//   C 16x16: lane n = lane&15, VGPR v -> M = v + 8*(lane>>4)
// ---------------------------------------------------------------------------

template<int sAm, int sAk, int sBk, int sBn, int M, int N, int K>
__device__ __forceinline__ v8f wmma_tile_f32(const float* __restrict__ ap,
                                             const float* __restrict__ bp,
                                             int m0, int n0, v8f acc)
{
  const int lane = threadIdx.x & 31;
  const int half = lane >> 4;
  const int ml   = lane & 15;
  const int am   = (M % 16 == 0) ? (m0 + ml) : min(m0 + ml, M - 1);
  const int bn   = (N % 16 == 0) ? (n0 + ml) : min(n0 + ml, N - 1);
  const float* apx = ap + am * sAm + (2 * half) * sAk;
  const float* bpx = bp + bn * sBn + (2 * half) * sBk;
  const int K4 = (K / 4) * 4;
  #pragma unroll 4
  for (int k0 = 0; k0 < K4; k0 += 4) {
    v2f a, b;
    a.x = apx[0];
    a.y = apx[sAk];
    b.x = bpx[0];
    b.y = bpx[sBk];
    acc = __builtin_amdgcn_wmma_f32_16x16x4_f32(false, a, false, b, (short)0, acc, false, false);
    apx += 4 * sAk;
    bpx += 4 * sBk;
  }
  if (K4 < K) {  // compile-time constant condition
    const int ka0 = K4 + 2 * half;
    const int ka1 = ka0 + 1;
    const float f0 = (ka0 < K) ? 1.0f : 0.0f;
    const float f1 = (ka1 < K) ? 1.0f : 0.0f;
    const int kc0 = min(ka0, K - 1);
    const int kc1 = min(ka1, K - 1);
    v2f a, b;
    a.x = ap[am * sAm + kc0 * sAk] * f0;
    a.y = ap[am * sAm + kc1 * sAk] * f1;
    b.x = bp[kc0 * sBk + bn * sBn];
    b.y = bp[kc1 * sBk + bn * sBn];
    acc = __builtin_amdgcn_wmma_f32_16x16x4_f32(false, a, false, b, (short)0, acc, false, false);
  }
  return acc;
}

// Two adjacent 16-wide N tiles per wave, sharing the A fragment (6 loads,
// 2 WMMA per k-step). Requires n0+32 <= N (N % 32 == 0 at all call sites).
template<int sAm, int sAk, int sBk, int sBn, int M, int N, int K>
__device__ __forceinline__ void wmma_tile2_f32(const float* __restrict__ ap,
                                               const float* __restrict__ bp,
                                               int m0, int n0, v8f& acc0, v8f& acc1)
{
  const int lane = threadIdx.x & 31;
  const int half = lane >> 4;
  const int ml   = lane & 15;
  const int am   = (M % 16 == 0) ? (m0 + ml) : min(m0 + ml, M - 1);
  const int bn   = n0 + ml;
  const float* apx  = ap + am * sAm + (2 * half) * sAk;
  const float* bpx0 = bp + bn * sBn + (2 * half) * sBk;
  const float* bpx1 = bpx0 + 16 * sBn;
  const int K4 = (K / 4) * 4;
  #pragma unroll 4
  for (int k0 = 0; k0 < K4; k0 += 4) {
    v2f a, b0, b1;
    a.x  = apx[0];
    a.y  = apx[sAk];
    b0.x = bpx0[0];
    b0.y = bpx0[sBk];
    b1.x = bpx1[0];
    b1.y = bpx1[sBk];
    acc0 = __builtin_amdgcn_wmma_f32_16x16x4_f32(false, a, false, b0, (short)0, acc0, false, false);
    acc1 = __builtin_amdgcn_wmma_f32_16x16x4_f32(false, a, false, b1, (short)0, acc1, false, false);
    apx  += 4 * sAk;
    bpx0 += 4 * sBk;
    bpx1 += 4 * sBk;
  }
  if (K4 < K) {
    const int ka0 = K4 + 2 * half;
    const int ka1 = ka0 + 1;
    const float f0 = (ka0 < K) ? 1.0f : 0.0f;
    const float f1 = (ka1 < K) ? 1.0f : 0.0f;
    const int kc0 = min(ka0, K - 1);
    const int kc1 = min(ka1, K - 1);
    v2f a, b0, b1;
    a.x  = ap[am * sAm + kc0 * sAk] * f0;
    a.y  = ap[am * sAm + kc1 * sAk] * f1;
    b0.x = bp[kc0 * sBk + bn * sBn];
    b0.y = bp[kc1 * sBk + bn * sBn];
    b1.x = bp[kc0 * sBk + (bn + 16) * sBn];
    b1.y = bp[kc1 * sBk + (bn + 16) * sBn];
    acc0 = __builtin_amdgcn_wmma_f32_16x16x4_f32(false, a, false, b0, (short)0, acc0, false, false);
    acc1 = __builtin_amdgcn_wmma_f32_16x16x4_f32(false, a, false, b1, (short)0, acc1, false, false);
  }
}

template<int M, int N>
__device__ __forceinline__ void wmma_store(float* cp, int sCm, int m0, int n0, v8f d)
{
  const int lane = threadIdx.x & 31;
  const int half = lane >> 4;
  const int n    = n0 + (lane & 15);
  if (N % 16 != 0 && n >= N) return;
  #pragma unroll
  for (int v = 0; v < 8; ++v) {
    const int m = m0 + v + 8 * half;
    if (M % 16 == 0 || m < M) cp[m * sCm + n] = d[v];
  }
}

// ---------------------------------------------------------------------------
// Stage kernels
// ---------------------------------------------------------------------------

// n_valid[b] = #(idxs>0); sep2[b] = idxs[b, n_valid-2]
__global__ void k_meta(const int* __restrict__ idxs, int* __restrict__ nvalid,
                       int* __restrict__ sep2)
{
  const int b = blockIdx.x;
  __shared__ int cnt;
  if (threadIdx.x == 0) cnt = 0;
  __syncthreads();
  const int t = threadIdx.x;
  if (t < T_ && idxs[b * T_ + t] > 0) atomicAdd(&cnt, 1);
  __syncthreads();
  if (threadIdx.x == 0) {
    const int nv = cnt;
    nvalid[b] = nv;
    sep2[b]   = idxs[b * T_ + (nv - 2)];
  }
}

// sent_emb (B,S,D) and arg_sub (B,A,D) gathers, masked
__global__ void k_gather(const float* __restrict__ emb, const int* __restrict__ idxs,
                         const int* __restrict__ nvalid, const int* __restrict__ sep2,
                         const int* __restrict__ blens,
                         float* __restrict__ sent, float* __restrict__ argsub)
{
  const int b = blockIdx.x;
  const int r = blockIdx.y;
  if (r < S_) {
    int ic = idxs[b * T_ + r];
    ic = min(max(ic, 0), L_ - 1);
    const float msk = (r < nvalid[b] - 2) ? 1.0f : 0.0f;
    const float* src = emb + ((long)b * L_ + ic) * D_;
    float* dst = sent + ((long)b * S_ + r) * D_;
    for (int d = threadIdx.x; d < D_; d += 256) dst[d] = src[d] * msk;
  } else {
    const int a   = r - S_;
    const int pos = sep2[b] + 1 + a;
    const int ic  = min(max(pos, 0), L_ - 1);
    const float msk = (pos < blens[b] - 1) ? 1.0f : 0.0f;
    const float* src = emb + ((long)b * L_ + ic) * D_;
    float* dst = argsub + ((long)b * A_ + a) * D_;
    for (int d = threadIdx.x; d < D_; d += 256) dst[d] = src[d] * msk;
  }
}

// trig[b,d] = sum_s sent*istrig / sum_s istrig
__global__ void k_trig(const float* __restrict__ sent, const float* __restrict__ istrig,
                       float* __restrict__ trig)
{
  const int b = blockIdx.x;
  float tsum = 0.0f;
  for (int s = 0; s < S_; ++s) tsum += istrig[b * S_ + s];
  const float inv = 1.0f / tsum;
  for (int d = threadIdx.x; d < D_; d += 256) {
    float acc = 0.0f;
    for (int s = 0; s < S_; ++s)
      acc += sent[((long)b * S_ + s) * D_ + d] * istrig[b * S_ + s];
    trig[b * D_ + d] = acc * inv;
  }
}

// out[b,e,d] = sum_s map[b,s,e] * src[b,s,d]   (ent0 and A_h2h) -- WMMA
__global__ void __launch_bounds__(32) k_bed(const float* __restrict__ src,
                                            const float* __restrict__ map,
                                            float* __restrict__ out)
{
  const int b = blockIdx.z, m0 = blockIdx.x * 16, n0 = blockIdx.y * 32;
  v8f a0 = {}, a1 = {};
  wmma_tile2_f32<1, E_, D_, 1, E_, D_, S_>(map + (long)b * S_ * E_,
                                           src + (long)b * S_ * D_, m0, n0, a0, a1);
  wmma_store<E_, D_>(out + (long)b * E_ * D_, D_, m0, n0, a0);
  wmma_store<E_, D_>(out + (long)b * E_ * D_, D_, m0, n0 + 16, a1);
}

// arg_emb[b,c,d] = sum_a argW[b,a,c] * argsub[b,a,d]  -- WMMA
__global__ void __launch_bounds__(32) k_argemb(const float* __restrict__ argW,
                                               const float* __restrict__ argsub,
                                               float* __restrict__ argemb)
{
  const int b = blockIdx.z, m0 = blockIdx.x * 16, n0 = blockIdx.y * 32;
  v8f a0 = {}, a1 = {};
  wmma_tile2_f32<1, A_, D_, 1, A_, D_, A_>(argW + (long)b * A_ * A_,
                                           argsub + (long)b * A_ * D_, m0, n0, a0, a1);
  wmma_store<A_, D_>(argemb + (long)b * A_ * D_, D_, m0, n0, a0);
  wmma_store<A_, D_>(argemb + (long)b * A_ * D_, D_, m0, n0 + 16, a1);
}

// entt = ent0 * trig (broadcast), nonemp[b,e] = (sum|ent0| != 0)
__global__ void k_entprep(const float* __restrict__ ent0, const float* __restrict__ trig,
                          float* __restrict__ entt, float* __restrict__ nonemp)
{
  const int b = blockIdx.x, e = blockIdx.y;
  __shared__ float red[256];
  float s = 0.0f;
  for (int d = threadIdx.x; d < D_; d += 256) {
    const float v = ent0[((long)b * E_ + e) * D_ + d];
    entt[((long)b * E_ + e) * D_ + d] = v * trig[b * D_ + d];
    s += fabsf(v);
  }
  red[threadIdx.x] = s;
  __syncthreads();
  for (int st = 128; st > 0; st >>= 1) {
    if (threadIdx.x < st) red[threadIdx.x] += red[threadIdx.x + st];
    __syncthreads();
  }
  if (threadIdx.x == 0) nonemp[b * E_ + e] = (red[0] != 0.0f) ? 1.0f : 0.0f;
}

// trigWt[b,n] = sum_k trig[b,k] * Wt[D+k, n]  (M=B single tile) -- WMMA
__global__ void __launch_bounds__(32) k_trigWt(const float* __restrict__ trig,
                                               const float* __restrict__ Wt,
                                               float* __restrict__ trigWt)
{
  const int n0 = blockIdx.y * 32;
  v8f a0 = {}, a1 = {};
  wmma_tile2_f32<D_, 1, D_, 1, B_, D_, D_>(trig, Wt + (long)D_ * D_, 0, n0, a0, a1);
  wmma_store<B_, D_>(trigWt, D_, 0, n0, a0);
  wmma_store<B_, D_>(trigWt, D_, 0, n0 + 16, a1);
}

// ent = (ent0@Wt0 + entt@Wt2 + trigWt + bt) * nonemp  -- WMMA + epilogue
__global__ void __launch_bounds__(32) k_ent(const float* __restrict__ ent0,
                                            const float* __restrict__ entt,
                                            const float* __restrict__ Wt,
                                            const float* __restrict__ trigWt,
                                            const float* __restrict__ bt,
                                            const float* __restrict__ nonemp,
                                            float* __restrict__ ent)
{
  const int b = blockIdx.z, m0 = blockIdx.x * 16, n0 = blockIdx.y * 32;
  v8f a0 = {}, a1 = {};
  wmma_tile2_f32<D_, 1, D_, 1, E_, D_, D_>(ent0 + (long)b * E_ * D_, Wt, m0, n0, a0, a1);
  wmma_tile2_f32<D_, 1, D_, 1, E_, D_, D_>(entt + (long)b * E_ * D_, Wt + 2L * D_ * D_,
                                           m0, n0, a0, a1);
  const int lane = threadIdx.x & 31, half = lane >> 4;
  #pragma unroll
  for (int t = 0; t < 2; ++t) {
    const int n = n0 + 16 * t + (lane & 15);
    const float add = trigWt[b * D_ + n] + bt[n];
    const v8f acc = t ? a1 : a0;
    #pragma unroll
    for (int v = 0; v < 8; ++v) {
      const int m = m0 + v + 8 * half;
      if (m < E_)
        ent[((long)b * E_ + m) * D_ + n] = (acc[v] + add) * nonemp[b * E_ + m];
    }
  }
}

// score_ta[b,e,a] = dot(ent[b,e], arg_emb[b,a]) / sqrt(D)  -- WMMA (B transposed)
__global__ void __launch_bounds__(32) k_score_ta(const float* __restrict__ ent,
                                                 const float* __restrict__ argemb,
                                                 float* __restrict__ sta)
{
  const int b = blockIdx.z, m0 = blockIdx.x * 16, n0 = blockIdx.y * 16;
  v8f acc = {};
  acc = wmma_tile_f32<D_, 1, 1, D_, E_, A_, D_>(ent + (long)b * E_ * D_,
                                                argemb + (long)b * A_ * D_, m0, n0, acc);
  const float sc = 0.03608439182435161f; // 1/sqrt(768)
  const int lane = threadIdx.x & 31, half = lane >> 4, n = n0 + (lane & 15);
  if (n < A_) {
    #pragma unroll
    for (int v = 0; v < 8; ++v) {
      const int m = m0 + v + 8 * half;
      if (m < E_) sta[((long)b * E_ + m) * A_ + n] = acc[v] * sc;
    }
  }
}

// rsum[b,e] = sum_a |sta|  ; csum[b,a] = sum_e |sta|
__global__ void k_norms(const float* __restrict__ sta, float* __restrict__ rsum,
                        float* __restrict__ csum)
{
  const int b = blockIdx.x, t = threadIdx.x;
  if (t < E_) {
    float s = 0.0f;
    for (int a = 0; a < A_; ++a) s += fabsf(sta[((long)b * E_ + t) * A_ + a]);
    rsum[b * E_ + t] = s;
  } else if (t >= 64 && (t - 64) < A_) {
    const int a = t - 64;
    float s = 0.0f;
    for (int e = 0; e < E_; ++e) s += fabsf(sta[((long)b * E_ + e) * A_ + a]);
    csum[b * A_ + a] = s;
  }
}

// tok_arg[b,e,d] = sum_a arg_emb[b,a,d] * sta[b,e,a] / max(rsum,eps)
__global__ void k_tokarg(const float* __restrict__ argemb, const float* __restrict__ sta,
                         const float* __restrict__ rsum, float* __restrict__ tokarg)
{
  const int b = blockIdx.x, e = blockIdx.y;
  const float inv = 1.0f / fmaxf(rsum[b * E_ + e], EPS_);
  for (int d = threadIdx.x; d < D_; d += 256) {
    float acc = 0.0f;
    for (int a = 0; a < A_; ++a)
      acc += argemb[((long)b * A_ + a) * D_ + d] * sta[((long)b * E_ + e) * A_ + a];
    tokarg[((long)b * E_ + e) * D_ + d] = acc * inv;
  }
}

// arg_tok[b,a,d] = sum_e ent[b,e,d] * sta[b,e,a] / max(csum,eps)
__global__ void k_argtok(const float* __restrict__ ent, const float* __restrict__ sta,
                         const float* __restrict__ csum, float* __restrict__ argtok)
{
  const int b = blockIdx.x, a = blockIdx.y;
  const float inv = 1.0f / fmaxf(csum[b * A_ + a], EPS_);
  for (int d = threadIdx.x; d < D_; d += 256) {
    float acc = 0.0f;
    for (int e = 0; e < E_; ++e)
      acc += ent[((long)b * E_ + e) * D_ + d] * sta[((long)b * E_ + e) * A_ + a];
    argtok[((long)b * A_ + a) * D_ + d] = acc * inv;
  }
}

// t2t[b,i,j]: gather 3 attention maps at (idx_i, idx_j), mask, row-normalize each, avg
__global__ void k_t2t(const float* __restrict__ att0, const float* __restrict__ att1,
                      const float* __restrict__ att2, const int* __restrict__ idxs,
                      const int* __restrict__ nvalid, float* __restrict__ t2t)
{
  const int b = blockIdx.x, i = blockIdx.y;
  __shared__ float r0[256], r1[256], r2[256];
  const int nv2 = nvalid[b] - 2;
  int ii = idxs[b * T_ + i];
  ii = min(max(ii, 0), L_ - 1);
  const float mi = (i < nv2) ? 1.0f : 0.0f;
  const int j = threadIdx.x;
  float v0 = 0.0f, v1 = 0.0f, v2 = 0.0f;
  if (j < S_) {
    int jj = idxs[b * T_ + j];
    jj = min(max(jj, 0), L_ - 1);
    const float m = mi * ((j < nv2) ? 1.0f : 0.0f);
    const long base = ((long)b * L_ + ii) * L_ + jj;
    v0 = att0[base] * m;
    v1 = att1[base] * m;
    v2 = att2[base] * m;
  }
  r0[j] = v0; r1[j] = v1; r2[j] = v2;
  __syncthreads();
  for (int st = 128; st > 0; st >>= 1) {
    if (threadIdx.x < st) {
      r0[threadIdx.x] += r0[threadIdx.x + st];
      r1[threadIdx.x] += r1[threadIdx.x + st];
      r2[threadIdx.x] += r2[threadIdx.x + st];
    }
    __syncthreads();
  }
  const float s0 = r0[0], s1 = r1[0], s2 = r2[0];
  if (j < S_) {
    const float out = (v0 / fmaxf(s0, EPS_) + v1 / fmaxf(s1, EPS_) +
                       v2 / fmaxf(s2, EPS_)) * (1.0f / 3.0f);
    t2t[((long)b * S_ + i) * S_ + j] = out;
  }
}

// h2h[b,i,d] = sum_j t2t[b,i,j] * sent[b,j,d]  -- WMMA (biggest GEMM)
__global__ void __launch_bounds__(32) k_h2h(const float* __restrict__ t2t,
                                            const float* __restrict__ sent,
                                            float* __restrict__ h2h)
{
  const int b = blockIdx.z, m0 = blockIdx.x * 16, n0 = blockIdx.y * 32;
  v8f a0 = {}, a1 = {};
  wmma_tile2_f32<S_, 1, D_, 1, S_, D_, S_>(t2t + (long)b * S_ * S_,
                                           sent + (long)b * S_ * D_, m0, n0, a0, a1);
  wmma_store<S_, D_>(h2h + (long)b * S_ * D_, D_, m0, n0, a0);
  wmma_store<S_, D_>(h2h + (long)b * S_ * D_, D_, m0, n0 + 16, a1);
}

// a2a[b,a,:] = softmax_c( dot(arg_emb[b,a], arg_emb[b,c]) )
__global__ void k_a2a(const float* __restrict__ argemb, float* __restrict__ a2a)
{
  const int b = blockIdx.x, a = blockIdx.y;
  __shared__ float g[A_];
  __shared__ float mx, sm;
  const int c = threadIdx.x;
  if (c < A_) {
    const float* x = argemb + ((long)b * A_ + a) * D_;
    const float* y = argemb + ((long)b * A_ + c) * D_;
    float s = 0.0f;
    for (int d = 0; d < D_; ++d) s += x[d] * y[d];
    g[c] = s;
  }
  __syncthreads();
  if (threadIdx.x == 0) {
    float m = g[0];
    for (int i = 1; i < A_; ++i) m = fmaxf(m, g[i]);
    float s = 0.0f;
    for (int i = 0; i < A_; ++i) s += expf(g[i] - m);
    mx = m; sm = s;
  }
  __syncthreads();
  if (c < A_) a2a[((long)b * A_ + a) * A_ + c] = expf(g[c] - mx) / sm;
}

// A_u2u[b,a,d] = sum_c a2a[b,a,c] * arg_emb[b,c,d]  -- WMMA
__global__ void __launch_bounds__(32) k_u2u(const float* __restrict__ a2a,
                                            const float* __restrict__ argemb,
                                            float* __restrict__ u2u)
{
  const int b = blockIdx.z, m0 = blockIdx.x * 16, n0 = blockIdx.y * 32;
  v8f a0 = {}, a1 = {};
  wmma_tile2_f32<A_, 1, D_, 1, A_, D_, A_>(a2a + (long)b * A_ * A_,
                                           argemb + (long)b * A_ * D_, m0, n0, a0, a1);
  wmma_store<A_, D_>(u2u + (long)b * A_ * D_, D_, m0, n0, a0);
  wmma_store<A_, D_>(u2u + (long)b * A_ * D_, D_, m0, n0 + 16, a1);
}

// U/V factor GEMMs: out[b,m,n] = sum of 3 (Ai @ W1_block_i) (+ optional bias)
template<int M>
__global__ void __launch_bounds__(32) k_UV(const float* __restrict__ A0,
                                           const float* __restrict__ A1,
                                           const float* __restrict__ A2,
                                           const float* __restrict__ W1,
                                           int blk0, int blk1, int blk2,
                                           const float* __restrict__ bias,
                                           float* __restrict__ out)
{
  const int b = blockIdx.z, m0 = blockIdx.x * 16, n0 = blockIdx.y * 32;
  v8f a0 = {}, a1 = {};
  wmma_tile2_f32<D_, 1, H_, 1, M, H_, D_>(A0 + (long)b * M * D_,
                                          W1 + (long)blk0 * D_ * H_, m0, n0, a0, a1);
  wmma_tile2_f32<D_, 1, H_, 1, M, H_, D_>(A1 + (long)b * M * D_,
                                          W1 + (long)blk1 * D_ * H_, m0, n0, a0, a1);
  wmma_tile2_f32<D_, 1, H_, 1, M, H_, D_>(A2 + (long)b * M * D_,
                                          W1 + (long)blk2 * D_ * H_, m0, n0, a0, a1);
  const int lane = threadIdx.x & 31, half = lane >> 4;
  #pragma unroll
  for (int t = 0; t < 2; ++t) {
    const int n = n0 + 16 * t + (lane & 15);
    const float add = bias ? bias[n] : 0.0f;
    const v8f acc = t ? a1 : a0;
    #pragma unroll
    for (int v = 0; v < 8; ++v) {
      const int m = m0 + v + 8 * half;
      if (m < M) out[((long)b * M + m) * H_ + n] = acc[v] + add;
    }
  }
}

// score[b,e,a] = W2 . gelu(U[b,e]+V[b,a]) + b2  (exact erf gelu)
__global__ void k_score(const float* __restrict__ U, const float* __restrict__ V,
                        const float* __restrict__ W2, const float* __restrict__ b2,
                        float* __restrict__ out)
{
  const int a = blockIdx.x, e = blockIdx.y, b = blockIdx.z;
  __shared__ float red[256];
  const float* u = U + ((long)b * E_ + e) * H_;
  const float* v = V + ((long)b * A_ + a) * H_;
  float acc = 0.0f;
  for (int n = threadIdx.x; n < H_; n += 256) {
    const float x = u[n] + v[n];
    const float h = 0.5f * x * (1.0f + erff(x * 0.70710678118654752f));
    acc += h * W2[n];
  }
  red[threadIdx.x] = acc;
  __syncthreads();
  for (int st = 128; st > 0; st >>= 1) {
    if (threadIdx.x < st) red[threadIdx.x] += red[threadIdx.x + st];
    __syncthreads();
  }
  if (threadIdx.x == 0) out[((long)b * E_ + e) * A_ + a] = red[0] + b2[0];
}

// ---------------------------------------------------------------------------
// Workspace layout (floats; first 32 slots reserved for int metadata)
// ---------------------------------------------------------------------------
#define WS_SENT    ((size_t)32)
#define WS_ARGSUB  (WS_SENT   + (size_t)B_ * S_ * D_)
#define WS_TRIG    (WS_ARGSUB + (size_t)B_ * A_ * D_)
#define WS_ENT0    (WS_TRIG   + (size_t)B_ * D_)
#define WS_ENTT    (WS_ENT0   + (size_t)B_ * E_ * D_)
#define WS_NONEMP  (WS_ENTT   + (size_t)B_ * E_ * D_)
#define WS_TRIGWT  (WS_NONEMP + (size_t)B_ * E_)
#define WS_ENT     (WS_TRIGWT + (size_t)B_ * D_)
#define WS_ARGEMB  (WS_ENT    + (size_t)B_ * E_ * D_)
#define WS_STA     (WS_ARGEMB + (size_t)B_ * A_ * D_)
#define WS_RSUM    (WS_STA    + (size_t)B_ * E_ * A_)
#define WS_CSUM    (WS_RSUM   + (size_t)B_ * E_)
#define WS_TOKARG  (WS_CSUM   + (size_t)B_ * A_)
#define WS_ARGTOK  (WS_TOKARG + (size_t)B_ * E_ * D_)
#define WS_T2T     (WS_ARGTOK + (size_t)B_ * A_ * D_)
#define WS_H2H     (WS_T2T    + (size_t)B_ * S_ * S_)
#define WS_AH2H    (WS_H2H    + (size_t)B_ * S_ * D_)
#define WS_A2A     (WS_AH2H   + (size_t)B_ * E_ * D_)
#define WS_U2U     (WS_A2A    + (size_t)B_ * A_ * A_)
#define WS_U       (WS_U2U    + (size_t)B_ * A_ * D_)
#define WS_V       (WS_U      + (size_t)B_ * E_ * H_)

extern "C" void kernel_launch(void* const* d_in, const int* in_sizes, int n_in,
                              void* d_out, int out_size, void* d_ws, size_t ws_size,
                              hipStream_t stream)
{
  (void)in_sizes; (void)n_in; (void)out_size; (void)ws_size;
  const float* emb    = (const float*)d_in[0];
  const float* istrig = (const float*)d_in[1];
  const float* argW   = (const float*)d_in[2];
  const float* entmap = (const float*)d_in[3];
  const float* att0   = (const float*)d_in[4];
  const float* att1   = (const float*)d_in[5];
  const float* att2   = (const float*)d_in[6];
  const float* Wt     = (const float*)d_in[7];
  const float* bt     = (const float*)d_in[8];
  const float* W1     = (const float*)d_in[9];
  const float* b1     = (const float*)d_in[10];
  const float* W2     = (const float*)d_in[11];
  const float* b2     = (const float*)d_in[12];
  const int*   idxs   = (const int*)d_in[13];
  const int*   blens  = (const int*)d_in[14];

  int*   meta   = (int*)d_ws;
  int*   nvalid = meta;
  int*   sep2   = meta + B_;
  float* f      = (float*)d_ws;

  float* sent   = f + WS_SENT;
  float* argsub = f + WS_ARGSUB;
  float* trig   = f + WS_TRIG;
  float* ent0   = f + WS_ENT0;
  float* entt   = f + WS_ENTT;
  float* nonemp = f + WS_NONEMP;
  float* trigWt = f + WS_TRIGWT;
  float* ent    = f + WS_ENT;
  float* argemb = f + WS_ARGEMB;
  float* sta    = f + WS_STA;
  float* rsum   = f + WS_RSUM;
  float* csum   = f + WS_CSUM;
  float* tokarg = f + WS_TOKARG;
  float* argtok = f + WS_ARGTOK;
  float* t2t    = f + WS_T2T;
  float* h2h    = f + WS_H2H;
  float* ah2h   = f + WS_AH2H;
  float* a2a    = f + WS_A2A;
  float* u2u    = f + WS_U2U;
  float* U      = f + WS_U;
  float* V      = f + WS_V;

  const int tE  = (E_ + 15) / 16;  // 2
  const int tA  = (A_ + 15) / 16;  // 3
  const int tS  = (S_ + 15) / 16;  // 16
  const int tD2 = D_ / 32;         // 24 (dual-N tiles)
  const int tH2 = H_ / 32;         // 16 (dual-N tiles)

  k_meta   <<<dim3(B_), dim3(256), 0, stream>>>(idxs, nvalid, sep2);
  k_gather <<<dim3(B_, S_ + A_), dim3(256), 0, stream>>>(emb, idxs, nvalid, sep2, blens,
                                                         sent, argsub);
  k_trig   <<<dim3(B_), dim3(256), 0, stream>>>(sent, istrig, trig);

  k_bed     <<<dim3(tE, tD2, B_), dim3(32), 0, stream>>>(sent, entmap, ent0);
  k_argemb  <<<dim3(tA, tD2, B_), dim3(32), 0, stream>>>(argW, argsub, argemb);
  k_entprep <<<dim3(B_, E_), dim3(256), 0, stream>>>(ent0, trig, entt, nonemp);
  k_trigWt  <<<dim3(1, tD2, 1), dim3(32), 0, stream>>>(trig, Wt, trigWt);
  k_ent     <<<dim3(tE, tD2, B_), dim3(32), 0, stream>>>(ent0, entt, Wt, trigWt, bt,
                                                         nonemp, ent);

  k_score_ta<<<dim3(tE, tA, B_), dim3(32), 0, stream>>>(ent, argemb, sta);
  k_norms   <<<dim3(B_), dim3(128), 0, stream>>>(sta, rsum, csum);
  k_tokarg  <<<dim3(B_, E_), dim3(256), 0, stream>>>(argemb, sta, rsum, tokarg);
  k_argtok  <<<dim3(B_, A_), dim3(256), 0, stream>>>(ent, sta, csum, argtok);

  k_t2t <<<dim3(B_, S_), dim3(256), 0, stream>>>(att0, att1, att2, idxs, nvalid, t2t);
  k_h2h <<<dim3(tS, tD2, B_), dim3(32), 0, stream>>>(t2t, sent, h2h);
  k_bed <<<dim3(tE, tD2, B_), dim3(32), 0, stream>>>(h2h, entmap, ah2h);

  k_a2a <<<dim3(B_, A_), dim3(64), 0, stream>>>(argemb, a2a);
  k_u2u <<<dim3(tA, tD2, B_), dim3(32), 0, stream>>>(a2a, argemb, u2u);

  // U = ent0@W1[0] + tok_arg@W1[2] + A_h2h@W1[3] + b1
  // V = arg_emb@W1[1] + arg_tok@W1[4] + A_u2u@W1[5]
  k_UV<E_> <<<dim3(tE, tH2, B_), dim3(32), 0, stream>>>(ent0, tokarg, ah2h, W1,
                                                        0, 2, 3, b1, U);
  k_UV<A_> <<<dim3(tA, tH2, B_), dim3(32), 0, stream>>>(argemb, argtok, u2u, W1,
                                                        1, 4, 5, (const float*)nullptr, V);

  k_score <<<dim3(A_, E_, B_), dim3(256), 0, stream>>>(U, V, W2, b2, (float*)d_out);
}